// OrthogonalLoss_6373731467993
// MI455X (gfx1250) — compile-verified
//
#include <hip/hip_runtime.h>

// B = 4096 rows, D = 4096 cols, f32 inputs. Whole problem is per-row
// sum-of-squares (4 matrices) + tiny combine -> 3 scalars.
// Bandwidth-bound: 256 MiB @ 23.3 TB/s ~= 11.5 us floor.
//
// Row sums run on the matrix pipe with V_WMMA_F32_16X16X4_F32:
// A = 16x4 tile of squared values, B = all-ones 4x16, C accumulates f32.
// Since B is all ones, the K-slot -> column mapping is irrelevant (every
// column just needs to be consumed exactly once), so each lane loads a
// contiguous b128 (4 floats) and feeds two WMMAs: (f0,f1) and (f2,f3),
// with the two lane-halves offset by 4 columns. 512 B per wave-load,
// 32 B contiguous per row. Loads are non-temporal: data is streamed once
// and the 256 MiB working set exceeds the 192 MB L2.

typedef __attribute__((ext_vector_type(2))) float v2f;
typedef __attribute__((ext_vector_type(4))) float v4f;
typedef __attribute__((ext_vector_type(8))) float v8f;

#define NB 4096
#define ND 4096
#define EPS_N 1e-12f

__global__ __launch_bounds__(256) void rowsq16_wmma(const float* __restrict__ x,
                                                    float* __restrict__ rowsum)
{
    __shared__ float part[8][16];

    const int rb   = blockIdx.x;      // 16-row block index
    const int tid  = threadIdx.x;     // 0..255 (8 waves of 32)
    const int wave = tid >> 5;        // 0..7  -> 512-col slice
    const int lane = tid & 31;
    const int half = lane >> 4;       // lane-half: cols +0 or +4
    const int lrow = lane & 15;       // M within 16-row tile

    // lane L  (<16): row rb*16+L, quad at cols k..k+3
    // lane L+16    : row rb*16+L, quad at cols k+4..k+7
    const float* __restrict__ rowp =
        x + ((size_t)rb * 16 + (size_t)lrow) * ND + wave * 512 + half * 4;

    v8f c0 = {};           // even-quad accumulator
    v8f c1 = {};           // odd-quad accumulator (breaks WMMA RAW chain)
    v2f ones;
    ones[0] = 1.0f;
    ones[1] = 1.0f;

#pragma unroll 4
    for (int k = 0; k < 512; k += 8) {
        v4f a = __builtin_nontemporal_load((const v4f*)(rowp + k)); // b128, 16B aligned
        v2f lo, hi;
        lo[0] = a[0] * a[0];
        lo[1] = a[1] * a[1];
        hi[0] = a[2] * a[2];
        hi[1] = a[3] * a[3];
        // D = A(16x4 squared) x ones(4x16) + C : every N column of D holds
        // the running row sum of M.
        c0 = __builtin_amdgcn_wmma_f32_16x16x4_f32(
                 false, lo, false, ones, (short)0, c0, false, false);
        c1 = __builtin_amdgcn_wmma_f32_16x16x4_f32(
                 false, hi, false, ones, (short)0, c1, false, false);
    }

    // D layout (32-bit 16x16 C/D): vgpr r, lanes 0-15 -> M=r, lanes 16-31 -> M=8+r.
    // Column N=0 lives in lane 0 (M=0..7) and lane 16 (M=8..15).
    if ((lane & 15) == 0) {
        const int rbase = half * 8;
#pragma unroll
        for (int r = 0; r < 8; ++r) part[wave][rbase + r] = c0[r] + c1[r];
    }
    __syncthreads();

    if (tid < 16) {
        float s = 0.0f;
#pragma unroll
        for (int w = 0; w < 8; ++w) s += part[w][tid];   // fixed order: deterministic
        rowsum[rb * 16 + tid] = s;
    }
}

__device__ __forceinline__ float qnorm(float s)
{
    // sum(normalize(x)^2) = s / max(sqrt(s), eps)^2   (s >= 0)
    float m = fmaxf(sqrtf(s), EPS_N);
    return s / (m * m);
}

__global__ __launch_bounds__(256) void finalize_loss(const float* __restrict__ rs,
                                                     float* __restrict__ out)
{
    __shared__ float sh1[256];
    __shared__ float sh2[256];
    const int tid = threadIdx.x;

    float a1 = 0.0f, a2 = 0.0f;
    for (int i = tid; i < NB; i += 256) {
        float sr  = rs[0 * NB + i];   // ||rna_i||^2
        float sw  = rs[1 * NB + i];   // ||wsi_i||^2
        float srw = rs[2 * NB + i];   // ||rna_wsi_i||^2
        float swr = rs[3 * NB + i];   // ||wsi_rna_i||^2
        a1 += qnorm(sr) * qnorm(sw);               // D1 term
        a2 += qnorm(swr + srw) * qnorm(sr + sw);   // D2 term (concat = sum of norms^2)
    }
    sh1[tid] = a1;
    sh2[tid] = a2;
    __syncthreads();
    for (int s = 128; s > 0; s >>= 1) {
        if (tid < s) {
            sh1[tid] += sh1[tid + s];
            sh2[tid] += sh2[tid + s];
        }
        __syncthreads();
    }
    if (tid == 0) {
        float d1 = sh1[0] / (float)NB;
        float d2 = sh2[0] / (float)NB;
        out[0] = 0.5f * d2 + 0.5f * d1;  // loss
        out[1] = d1;                     // loss_D1
        out[2] = d2;                     // loss_D2
    }
}

extern "C" void kernel_launch(void* const* d_in, const int* in_sizes, int n_in,
                              void* d_out, int out_size, void* d_ws, size_t ws_size,
                              hipStream_t stream)
{
    (void)in_sizes; (void)n_in; (void)out_size; (void)ws_size;

    const float* rna    = (const float*)d_in[0];  // uni_repr_rna
    const float* wsi    = (const float*)d_in[1];  // uni_repr_wsi
    const float* rnawsi = (const float*)d_in[2];  // uni_repr_rna_wsi
    const float* wsirna = (const float*)d_in[3];  // uni_repr_wsi_rna

    float* rs = (float*)d_ws;  // 4 * NB floats = 64 KB, write-before-read

    dim3 grid(NB / 16), block(256);
    rowsq16_wmma<<<grid, block, 0, stream>>>(rna,    rs + 0 * NB);
    rowsq16_wmma<<<grid, block, 0, stream>>>(wsi,    rs + 1 * NB);
    rowsq16_wmma<<<grid, block, 0, stream>>>(rnawsi, rs + 2 * NB);
    rowsq16_wmma<<<grid, block, 0, stream>>>(wsirna, rs + 3 * NB);
    finalize_loss<<<1, 256, 0, stream>>>(rs, (float*)d_out);
}